// AttentionLayer_68384469287426
// MI455X (gfx1250) — compile-verified
//
#include <hip/hip_runtime.h>
#include <hip/hip_bf16.h>
#include <math.h>

#define N_NODES   8192
#define HIDDEN    256
#define NUM_HEADS 4
#define HEAD_DIM  64

typedef float v2f __attribute__((ext_vector_type(2)));
typedef float v8f __attribute__((ext_vector_type(8)));

// ---------------------------------------------------------------------------
// FP32 WMMA GEMM:  C[M x 256] = A[M x 256] * W[256 x 256] + bias
// One wave computes a 16x16 tile of C via V_WMMA_F32_16X16X4_F32 (K loop of 64).
// A-frag (16x4):  lanes 0-15 -> M=lane, K={0,1}; lanes 16-31 -> M=lane-16, K={2,3}
// B-frag (4x16):  lanes 0-15 -> N=lane, K={0,1}; lanes 16-31 -> N=lane-16, K={2,3}
// C/D   (16x16):  VGPR r: lanes 0-15 -> M=r, N=lane; lanes 16-31 -> M=r+8
// ---------------------------------------------------------------------------
__global__ void gemm256_wmma_f32(const float* __restrict__ A,
                                 const float* __restrict__ W,
                                 const float* __restrict__ bias,
                                 float* __restrict__ C, int M) {
    const int wave = (blockIdx.x * blockDim.x + threadIdx.x) >> 5;
    const int lane = threadIdx.x & 31;
    const int ntn  = HIDDEN / 16;               // 16 N-tiles
    const int tm   = wave / ntn;
    const int tn   = wave % ntn;
    const int row0 = tm * 16;
    const int col0 = tn * 16;
    if (row0 >= M) return;                       // wave-uniform: EXEC stays all-1s

    const int lm = lane & 15;                    // M index (A) / N index (B,C)
    const int kh = (lane >> 4) << 1;             // K sub-offset: 0 or 2

    const float* arow = A + (size_t)(row0 + lm) * HIDDEN;
    v8f c = {};
    for (int k = 0; k < HIDDEN; k += 4) {
        v2f a, b;
        a.x = arow[k + kh];
        a.y = arow[k + kh + 1];
        b.x = W[(size_t)(k + kh) * HIDDEN + col0 + lm];
        b.y = W[(size_t)(k + kh + 1) * HIDDEN + col0 + lm];
        c = __builtin_amdgcn_wmma_f32_16x16x4_f32(false, a, false, b,
                                                  (short)0, c, false, false);
    }
    const float bv  = bias[col0 + lm];
    const int   cm0 = row0 + ((lane >> 4) << 3); // +8 rows for upper half-wave
    for (int r = 0; r < 8; ++r)
        C[(size_t)(cm0 + r) * HIDDEN + col0 + lm] = c[r] + bv;
}

// ---------------------------------------------------------------------------
// CSR construction over edge_index (2 x E), int32, row 0 = src (softmax rows)
// ---------------------------------------------------------------------------
__global__ void zero_i32(int* p, int n) {
    int i = blockIdx.x * blockDim.x + threadIdx.x;
    if (i < n) p[i] = 0;
}

__global__ void edge_count(const int* __restrict__ edge, int E,
                           int* __restrict__ counts) {
    int e = blockIdx.x * blockDim.x + threadIdx.x;
    if (e < E) atomicAdd(&counts[edge[e]], 1);
}

// exclusive scan of 8192 counts -> row_start[0..8192]; single block, 256 thr
__global__ void scan8192(const int* __restrict__ counts,
                         int* __restrict__ row_start) {
    __shared__ int sums[256];
    const int t = threadIdx.x;
    const int per = N_NODES / 256;               // 32
    const int base = t * per;
    int local[per];
    int s = 0;
    for (int i = 0; i < per; ++i) { local[i] = s; s += counts[base + i]; }
    sums[t] = s;
    __syncthreads();
    for (int off = 1; off < 256; off <<= 1) {    // Hillis-Steele inclusive
        int v = (t >= off) ? sums[t - off] : 0;
        __syncthreads();
        sums[t] += v;
        __syncthreads();
    }
    const int woff = (t == 0) ? 0 : sums[t - 1];
    for (int i = 0; i < per; ++i) row_start[base + i] = woff + local[i];
    if (t == 255) row_start[N_NODES] = sums[255];
}

__global__ void edge_scatter(const int* __restrict__ edge, int E,
                             const int* __restrict__ row_start,
                             int* __restrict__ cursor,
                             int* __restrict__ colidx) {
    int e = blockIdx.x * blockDim.x + threadIdx.x;
    if (e < E) {
        int s = edge[e];
        int d = edge[E + e];
        int pos = atomicAdd(&cursor[s], 1);
        colidx[row_start[s] + pos] = d;
    }
}

// sort each row's column list (determinism + adjacent-duplicate dedup)
__global__ void sort_rows(int* __restrict__ colidx,
                          const int* __restrict__ row_start) {
    int i = blockIdx.x * blockDim.x + threadIdx.x;
    if (i >= N_NODES) return;
    int s = row_start[i], e = row_start[i + 1];
    for (int a = s + 1; a < e; ++a) {
        int key = colidx[a];
        int b = a - 1;
        while (b >= s && colidx[b] > key) { colidx[b + 1] = colidx[b]; --b; }
        colidx[b + 1] = key;
    }
}

// ---------------------------------------------------------------------------
// Sparse masked attention with online softmax.
// exp(score - 1e9 - max) == 0.0f exactly in fp32 for non-edge entries, so
// per-row softmax over the (deduped) edge set is numerically identical to the
// dense masked reference. Empty rows: the -1e9 shift cancels -> dense softmax.
// One wave per (node, head); lane owns dims {lane, lane+32} of HEAD_DIM=64.
// ---------------------------------------------------------------------------
__global__ void sparse_attention(const float* __restrict__ Q,
                                 const float* __restrict__ K,
                                 const float* __restrict__ V,
                                 const int* __restrict__ row_start,
                                 const int* __restrict__ colidx,
                                 float* __restrict__ Ctx) {
    const int i    = blockIdx.x;
    const int h    = threadIdx.x >> 5;           // 4 waves = 4 heads
    const int lane = threadIdx.x & 31;
    const int off  = h * HEAD_DIM;

    const float q0 = Q[(size_t)i * HIDDEN + off + lane];
    const float q1 = Q[(size_t)i * HIDDEN + off + lane + 32];

    float m = -INFINITY, l = 0.0f, acc0 = 0.0f, acc1 = 0.0f;
    const float scale_s = 0.125f;                // 1/sqrt(64)

    const int s = row_start[i], e = row_start[i + 1];
    if (e > s) {
        int prev = -1;
        for (int ei = s; ei < e; ++ei) {
            const int j = colidx[ei];
            if (j == prev) continue;             // dedup (sorted)
            prev = j;
            const float* kj = K + (size_t)j * HIDDEN + off;
            float part = q0 * kj[lane] + q1 * kj[lane + 32];
            for (int o = 16; o > 0; o >>= 1)
                part += __shfl_xor(part, o, 32);
            const float sc   = part * scale_s;
            const float mnew = fmaxf(m, sc);
            const float r    = __expf(m - mnew); // exp(-inf)=0 on first edge
            const float w    = __expf(sc - mnew);
            const float* vj  = V + (size_t)j * HIDDEN + off;
            l    = l * r + w;
            acc0 = acc0 * r + w * vj[lane];
            acc1 = acc1 * r + w * vj[lane + 32];
            m = mnew;
        }
    } else {
        // empty row: constant -1e9 cancels -> dense softmax over all keys
        for (int j = 0; j < N_NODES; ++j) {
            const float* kj = K + (size_t)j * HIDDEN + off;
            float part = q0 * kj[lane] + q1 * kj[lane + 32];
            for (int o = 16; o > 0; o >>= 1)
                part += __shfl_xor(part, o, 32);
            const float sc   = part * scale_s;
            const float mnew = fmaxf(m, sc);
            const float r    = __expf(m - mnew);
            const float w    = __expf(sc - mnew);
            const float* vj  = V + (size_t)j * HIDDEN + off;
            l    = l * r + w;
            acc0 = acc0 * r + w * vj[lane];
            acc1 = acc1 * r + w * vj[lane + 32];
            m = mnew;
        }
    }
    const float inv = 1.0f / l;
    Ctx[(size_t)i * HIDDEN + off + lane]      = acc0 * inv;
    Ctx[(size_t)i * HIDDEN + off + lane + 32] = acc1 * inv;
}

// ---------------------------------------------------------------------------
extern "C" void kernel_launch(void* const* d_in, const int* in_sizes, int n_in,
                              void* d_out, int out_size, void* d_ws, size_t ws_size,
                              hipStream_t stream) {
    const float* X   = (const float*)d_in[0];
    const float* Wq  = (const float*)d_in[1];
    const float* bq  = (const float*)d_in[2];
    const float* Wk  = (const float*)d_in[3];
    const float* bk  = (const float*)d_in[4];
    const float* Wv  = (const float*)d_in[5];
    const float* bv  = (const float*)d_in[6];
    const float* Wo  = (const float*)d_in[7];
    const float* bo  = (const float*)d_in[8];
    const int*   edge = (const int*)d_in[9];
    const int    E    = in_sizes[9] / 2;

    // workspace layout
    const size_t NM = (size_t)N_NODES * HIDDEN;
    float* Q   = (float*)d_ws;
    float* K   = Q + NM;
    float* V   = K + NM;
    float* Ctx = V + NM;
    int* counts    = (int*)(Ctx + NM);
    int* cursor    = counts + N_NODES;
    int* row_start = cursor + N_NODES;           // N_NODES + 1 entries
    int* colidx    = row_start + N_NODES + 1;    // E entries

    // --- QKV projections (fp32 WMMA GEMM) ---
    const int waves  = (N_NODES / 16) * (HIDDEN / 16);  // 8192 waves
    const int blocks = waves * 32 / 256;
    gemm256_wmma_f32<<<blocks, 256, 0, stream>>>(X, Wq, bq, Q, N_NODES);
    gemm256_wmma_f32<<<blocks, 256, 0, stream>>>(X, Wk, bk, K, N_NODES);
    gemm256_wmma_f32<<<blocks, 256, 0, stream>>>(X, Wv, bv, V, N_NODES);

    // --- CSR build ---
    zero_i32<<<(2 * N_NODES + 255) / 256, 256, 0, stream>>>(counts, 2 * N_NODES);
    edge_count<<<(E + 255) / 256, 256, 0, stream>>>(edge, E, counts);
    scan8192<<<1, 256, 0, stream>>>(counts, row_start);
    edge_scatter<<<(E + 255) / 256, 256, 0, stream>>>(edge, E, row_start, cursor, colidx);
    sort_rows<<<(N_NODES + 255) / 256, 256, 0, stream>>>(colidx, row_start);

    // --- sparse masked attention ---
    sparse_attention<<<N_NODES, NUM_HEADS * 32, 0, stream>>>(Q, K, V, row_start,
                                                             colidx, Ctx);

    // --- output projection ---
    gemm256_wmma_f32<<<blocks, 256, 0, stream>>>(Ctx, Wo, bo, (float*)d_out, N_NODES);
}